// MultiHeadAttentionQuantum_65481071395260
// MI455X (gfx1250) — compile-verified
//
#include <hip/hip_runtime.h>
#include <math.h>

typedef __attribute__((ext_vector_type(2))) float v2f;
typedef __attribute__((ext_vector_type(8))) float v8f;

#define SEQ   4096
#define BATCH 16
#define NTOK  (BATCH * SEQ)

// ---------------- Phase 1: quantum features (analytic form) ----------------
// q[0] = prod_{w=1..7} cos(x_w + th_w); q[k] = prod_{w=0..k} cos(x_w + th_w)
__global__ void qfeat_kernel(const float* __restrict__ x,
                             const float* __restrict__ theta,
                             float* __restrict__ q) {
    int n = blockIdx.x * blockDim.x + threadIdx.x;
    if (n >= NTOK) return;
    float c[8];
#pragma unroll
    for (int w = 0; w < 8; ++w) c[w] = cosf(x[n * 8 + w] + theta[w]);
    float out[8];
    out[0] = c[1] * c[2] * c[3] * c[4] * c[5] * c[6] * c[7];
    float p = c[0];
#pragma unroll
    for (int k = 1; k < 8; ++k) { p *= c[k]; out[k] = p; }
    float4* dst = (float4*)(q + n * 8);
    dst[0] = make_float4(out[0], out[1], out[2], out[3]);
    dst[1] = make_float4(out[4], out[5], out[6], out[7]);
}

// ---------------- Phase 2: attention via f32 WMMA ----------------
// One wave handles 16 query rows; loops over 16-column tiles of all 4096 keys.
// QK^T: 2x V_WMMA_F32_16X16X4_F32 (K=8). |q|<=1 bounds scores -> raw v_exp_f32
// is safe and no running-max pass is needed. Softmax scale pre-folded into the
// Q fragments. P staged through LDS (C-layout -> A-layout); V tile staged
// transposed with an extra all-ones row so the P.V WMMA's column n=8 yields
// the softmax denominator directly in the accumulator layout.
#define PW_STRIDE 18
#define WAVE_LDS  450   // P: 16*18 + Vt: 9*18

__global__ __launch_bounds__(128) void attn_kernel(const float* __restrict__ q,
                                                   float* __restrict__ attn_out) {
    __shared__ __align__(16) float lds[4 * WAVE_LDS];
    const int lane = threadIdx.x & 31;
    const int wid  = threadIdx.x >> 5;
    const int hi   = lane >> 4;   // half-wave select
    const int tl   = lane & 15;
    const int batch = blockIdx.y;
    const int row0  = blockIdx.x * 64 + wid * 16;
    const float* qb = q + (size_t)batch * SEQ * 8;

    float* Pw = lds + wid * WAVE_LDS; // P tile, stride 18 (conflict-free, 8B-aligned pairs)
    float* Vw = Pw + 288;             // V^T tile (9 rows x 18), row 8 = ones

    if (lane < 16) Vw[8 * PW_STRIDE + lane] = 1.0f; // ones column, written once

    // Q tile as WMMA A fragments (M=row=lane%16, K = v + 2*hi), pre-scaled
    const float SC = 1.4426950408889634f * 0.70710678118654752f; // log2(e)/sqrt(2)
    const int rg = row0 + tl;
    v2f a0 = *(const v2f*)(qb + rg * 8 + 2 * hi);
    v2f a1 = *(const v2f*)(qb + rg * 8 + 4 + 2 * hi);
    a0 *= SC;
    a1 *= SC;

    v8f oacc = {};
    const int ec = (tl < 8) ? tl : 8; // B-fragment N index; n=8 -> ones row (denom)

    for (int j = 0; j < SEQ; j += 16) {
        // K tile as B fragments (N=token=lane%16, K = v + 2*hi)
        const float* kt = qb + (size_t)(j + tl) * 8;
        v2f b0 = *(const v2f*)(kt + 2 * hi);
        v2f b1 = *(const v2f*)(kt + 4 + 2 * hi);

        v8f s = {};
        s = __builtin_amdgcn_wmma_f32_16x16x4_f32(false, a0, false, b0, (short)0, s, false, false);
        s = __builtin_amdgcn_wmma_f32_16x16x4_f32(false, a1, false, b1, (short)0, s, false, false);

        // P = exp2(scaled score); store in row-major [row][col] (C layout: row=v+8hi, col=tl)
#pragma unroll
        for (int v = 0; v < 8; ++v) {
            float pe = __builtin_amdgcn_exp2f(s[v]);
            Pw[(v + 8 * hi) * PW_STRIDE + tl] = pe;
        }
        // stage V tile transposed: Vt[e][t] (rows 0..7 only; row 8 stays ones)
        Vw[(2 * hi)     * PW_STRIDE + tl] = b0.x;
        Vw[(2 * hi + 1) * PW_STRIDE + tl] = b0.y;
        Vw[(4 + 2 * hi) * PW_STRIDE + tl] = b1.x;
        Vw[(5 + 2 * hi) * PW_STRIDE + tl] = b1.y;
        __syncthreads();

        // read P back in A layout, V^T in B layout; accumulate P.V (+denom in col 8)
#pragma unroll
        for (int f = 0; f < 4; ++f) {
            v2f pa = *(const v2f*)(Pw + tl * PW_STRIDE + 4 * f + 2 * hi);
            v2f vb = *(const v2f*)(Vw + ec * PW_STRIDE + 4 * f + 2 * hi);
            oacc = __builtin_amdgcn_wmma_f32_16x16x4_f32(false, pa, false, vb, (short)0, oacc, false, false);
        }
        __syncthreads();
    }

    // Denominators live in column n=8 (lanes 8 and 24), same row layout as O.
    float dv[8];
#pragma unroll
    for (int v = 0; v < 8; ++v) dv[v] = __shfl(oacc[v], 8 + 16 * hi, 32);

    // O layout: lane -> col e = lane%16 (valid e<8), rows r = v + 8*hi
    if (tl < 8) {
        const int e = tl;
#pragma unroll
        for (int v = 0; v < 8; ++v) {
            int r = row0 + v + 8 * hi;
            attn_out[((size_t)batch * SEQ + r) * 8 + e] = oacc[v] / dv[v];
        }
    }
}

// ---------------- Phase 3: head-transpose scramble + output projection ----------------
// out[b,s,e] = bias[e] + sum_f W[e,f] * attn[b, (s%512)*8+f, s/512]
__global__ void combine_kernel(const float* __restrict__ attn,
                               const float* __restrict__ W,
                               const float* __restrict__ bias,
                               float* __restrict__ out) {
    int g = blockIdx.x * blockDim.x + threadIdx.x;
    if (g >= NTOK) return;
    int b = g >> 12;
    int s = g & 4095;
    int esrc  = s >> 9;
    int rbase = (s & 511) * 8;
    const float* ab = attn + ((size_t)b * SEQ + rbase) * 8 + esrc;
    float a[8];
#pragma unroll
    for (int f = 0; f < 8; ++f) a[f] = ab[f * 8];
#pragma unroll
    for (int e = 0; e < 8; ++e) {
        float acc = bias[e];
#pragma unroll
        for (int f = 0; f < 8; ++f) acc += W[e * 8 + f] * a[f];
        out[(size_t)g * 8 + e] = acc;
    }
}

extern "C" void kernel_launch(void* const* d_in, const int* in_sizes, int n_in,
                              void* d_out, int out_size, void* d_ws, size_t ws_size,
                              hipStream_t stream) {
    const float* x     = (const float*)d_in[0];
    const float* theta = (const float*)d_in[1];
    const float* w     = (const float*)d_in[2];
    const float* bias  = (const float*)d_in[3];

    float* q    = (float*)d_ws;              // NTOK*8 floats (2 MB)
    float* attn = q + (size_t)NTOK * 8 + 64; // NTOK*8 floats (2 MB), padded offset

    qfeat_kernel<<<NTOK / 256, 256, 0, stream>>>(x, theta, q);
    attn_kernel<<<dim3(64, BATCH), 128, 0, stream>>>(q, attn);
    combine_kernel<<<NTOK / 256, 256, 0, stream>>>(attn, w, bias, (float*)d_out);
}